// Classifier_58188216926982
// MI455X (gfx1250) — compile-verified
//
#include <hip/hip_runtime.h>

typedef __attribute__((ext_vector_type(2))) float v2f;
typedef __attribute__((ext_vector_type(8))) float v8f;

#define NHID 256
#define EPSF 1e-7f
#define MAXNF 0.99999f   // 1 - 1e-5

__device__ __forceinline__ v8f wmma_f32x4(v2f a, v2f b, v8f c) {
  // D = A(16x4 f32) * B(4x16 f32) + C(16x16 f32)
  return __builtin_amdgcn_wmma_f32_16x16x4_f32(false, a, false, b, (short)0, c,
                                               false, false);
}

__device__ __forceinline__ float wred32(float v) {
#pragma unroll
  for (int m = 16; m > 0; m >>= 1) v += __shfl_xor(v, m, 32);
  return v;
}

// tanh(mxn/xn * atanh(xn)) / mxn for (already-scaled) squared norms
__device__ __forceinline__ float hyp_scale(float x2, float m2) {
  float xn  = fminf(fmaxf(sqrtf(x2), EPSF), MAXNF);
  float mxn = fmaxf(sqrtf(m2), EPSF);
  return tanhf(mxn / xn * atanhf(xn)) / mxn;
}

__device__ __forceinline__ void load8(const float* p, float s, float* d) {
  const float4* g = (const float4*)p;
  float4 a = g[0], b = g[1];
  d[0] = a.x * s; d[1] = a.y * s; d[2] = a.z * s; d[3] = a.w * s;
  d[4] = b.x * s; d[5] = b.y * s; d[6] = b.z * s; d[7] = b.w * s;
}

__device__ __forceinline__ void store8(float* p, const float* d, float s) {
  float4 a, b;
  a.x = d[0] * s; a.y = d[1] * s; a.z = d[2] * s; a.w = d[3] * s;
  b.x = d[4] * s; b.y = d[5] * s; b.z = d[6] * s; b.w = d[7] * s;
  float4* g = (float4*)p;
  g[0] = a; g[1] = b;
}

// ---------------------------------------------------------------------------
// GEMM + mobius_matvec epilogue. Block = 32 rows x 256 cols (full width).
// 8 waves: wave w -> rows [16*(w/4)..), cols [64*(w%4)..), 4 WMMA accums.
// FULL_EPI additionally applies: rst = 0.25*mv; mobius_add(rst, bias);
//                                expmap0(relu(logmap0(rst)))   (layer 2)
// May run in-place (X == OUT): every block only reads its own 32 rows and
// writes them after barriers.
// ---------------------------------------------------------------------------
template <int KDIM, bool FULL_EPI>
__global__ __launch_bounds__(256) void k_mm(const float* X,
                                            const float* __restrict__ W,
                                            const float* __restrict__ bias,
                                            float* OUT, float inScale) {
  __shared__ float smx[32][NHID + 8];
  __shared__ float part[32][8];
  __shared__ float rx2[32], rm2[32], rdot[32], rv2[32];
  __shared__ float sb2s;

  const int t = threadIdx.x;
  const int wave = t >> 5, lane = t & 31;
  const int tile_m = wave >> 2;          // 0..1
  const int n0 = (wave & 3) << 6;        // 0,64,128,192
  const int l15 = lane & 15;
  const int kk = (lane >> 4) << 1;       // 0 or 2 (A/B fragment K-sublane)
  const int brow = blockIdx.x << 5;

  v8f acc0 = {}, acc1 = {}, acc2 = {}, acc3 = {};
  const float* Ar = X + (size_t)(brow + tile_m * 16 + l15) * KDIM;
  for (int k0 = 0; k0 < KDIM; k0 += 4) {
    v2f a;
    a.x = Ar[k0 + kk];
    a.y = Ar[k0 + kk + 1];
    const float* Bp = W + (size_t)(k0 + kk) * NHID + n0 + l15;
    v2f b;
    b.x = Bp[0];  b.y = Bp[NHID];      acc0 = wmma_f32x4(a, b, acc0);
    b.x = Bp[16]; b.y = Bp[NHID + 16]; acc1 = wmma_f32x4(a, b, acc1);
    b.x = Bp[32]; b.y = Bp[NHID + 32]; acc2 = wmma_f32x4(a, b, acc2);
    b.x = Bp[48]; b.y = Bp[NHID + 48]; acc3 = wmma_f32x4(a, b, acc3);
  }
  {
    // C/D layout: lanes 0-15 -> M = v, lanes 16-31 -> M = v+8; N = lane&15
    const int mb = tile_m * 16 + ((lane >> 4) ? 8 : 0);
#pragma unroll
    for (int v = 0; v < 8; ++v) {
      smx[mb + v][n0 + 0 + l15]  = acc0[v];
      smx[mb + v][n0 + 16 + l15] = acc1[v];
      smx[mb + v][n0 + 32 + l15] = acc2[v];
      smx[mb + v][n0 + 48 + l15] = acc3[v];
    }
  }
  __syncthreads();

  const int r = t >> 3, p = t & 7;  // 8 threads per output row
  // |x_row|^2 (raw, unscaled)
  float sx = 0.f;
  {
    const int seg = KDIM / 8;
    const float* xr = X + (size_t)(brow + r) * KDIM + p * seg;
    for (int j = 0; j < seg; j += 4) {
      float4 v = *(const float4*)(xr + j);
      sx += v.x * v.x + v.y * v.y + v.z * v.z + v.w * v.w;
    }
  }
  // |mx_row|^2 and dot(mx_row, bias) (raw)
  float sm = 0.f, sd = 0.f;
  {
    const float* mr = &smx[r][p * 32];
#pragma unroll
    for (int j = 0; j < 32; ++j) {
      float m = mr[j];
      sm += m * m;
      if (FULL_EPI) sd += m * bias[p * 32 + j];
    }
  }
  part[r][p] = sx;
  __syncthreads();
  if (t < 32) { float s = 0.f; for (int i = 0; i < 8; ++i) s += part[t][i]; rx2[t] = s; }
  __syncthreads();
  part[r][p] = sm;
  __syncthreads();
  if (t < 32) { float s = 0.f; for (int i = 0; i < 8; ++i) s += part[t][i]; rm2[t] = s; }
  __syncthreads();
  if (FULL_EPI) {
    part[r][p] = sd;
    __syncthreads();
    if (t < 32) { float s = 0.f; for (int i = 0; i < 8; ++i) s += part[t][i]; rdot[t] = s; }
    __syncthreads();
    part[r][p] = bias[t] * bias[t];
    __syncthreads();
    if (t == 0) {
      float s = 0.f;
      for (int i = 0; i < 32; ++i)
        for (int j = 0; j < 8; ++j) s += part[i][j];
      sb2s = s;
    }
    __syncthreads();
  }

  // mobius_matvec scale: result = cmv * mx_raw
  const float x2s = inScale * inScale * rx2[r];
  const float m2s = inScale * inScale * rm2[r];
  const float cmv = hyp_scale(x2s, m2s) * inScale;
  float* orow = OUT + (size_t)(brow + r) * NHID + p * 32;
  float* mrw = &smx[r][p * 32];
  if (!FULL_EPI) {
#pragma unroll
    for (int j = 0; j < 32; j += 4) {
      float4 o;
      o.x = cmv * mrw[j + 0]; o.y = cmv * mrw[j + 1];
      o.z = cmv * mrw[j + 2]; o.w = cmv * mrw[j + 3];
      *(float4*)(orow + j) = o;
    }
  } else {
    const float cf = 0.25f * cmv;              // rst = cf * mx_raw
    const float x2r = cf * cf * rm2[r];
    const float xy = cf * rdot[r];
    const float y2 = sb2s;
    const float den = fmaxf(1.f + 2.f * xy + x2r * y2, EPSF);
    const float ca = (1.f + 2.f * xy + y2) / den;
    const float cb = (1.f - x2r) / den;
    const float zn2 = ca * ca * x2r + 2.f * ca * cb * xy + cb * cb * y2;
    const float zn = fminf(fmaxf(sqrtf(zn2), EPSF), MAXNF);
    const float us = atanhf(zn) / zn;          // logmap0 scale
    const float acf = ca * cf;
    float sv = 0.f;
#pragma unroll
    for (int j = 0; j < 32; ++j) {
      float z = acf * mrw[j] + cb * bias[p * 32 + j];
      float uu = fmaxf(us * z, 0.f);           // relu in tangent space
      mrw[j] = uu;
      sv += uu * uu;
    }
    part[r][p] = sv;
    __syncthreads();
    if (t < 32) { float s = 0.f; for (int i = 0; i < 8; ++i) s += part[t][i]; rv2[t] = s; }
    __syncthreads();
    const float vn = fmaxf(sqrtf(rv2[r]), EPSF);
    const float fs = tanhf(vn) / vn;           // expmap0 scale
#pragma unroll
    for (int j = 0; j < 32; j += 4) {
      float4 o;
      o.x = fs * mrw[j + 0]; o.y = fs * mrw[j + 1];
      o.z = fs * mrw[j + 2]; o.w = fs * mrw[j + 3];
      *(float4*)(orow + j) = o;
    }
  }
}

// ---------------------------------------------------------------------------
// Mobius aggregate: one wave32 per node (8 floats/lane = 1KB coalesced row
// gathers, L2-resident). Serial chain of 15 mobius_adds (true scan dep).
// doAct: rst=0.25*acc; mobius_add(rst, bias); expmap0(relu(logmap0)).
// ---------------------------------------------------------------------------
__global__ __launch_bounds__(256) void k_agg(const float* __restrict__ F,
                                             const int* __restrict__ idx,
                                             const float* __restrict__ bias,
                                             float* __restrict__ OUT,
                                             float inScale, int doAct) {
  const int t = threadIdx.x;
  const int wave = t >> 5, lane = t & 31;
  const int node = (blockIdx.x << 3) + wave;
  const int* srow = idx + (size_t)node * 16;
  const int joff = lane << 3;

  float x[8];
  load8(F + (size_t)srow[0] * NHID + joff, inScale, x);
  float px = 0.f;
#pragma unroll
  for (int j = 0; j < 8; ++j) px += x[j] * x[j];
  float x2 = wred32(px);

  for (int k = 1; k < 16; ++k) {
    if (k < 15) __builtin_prefetch(F + (size_t)srow[k + 1] * NHID + joff, 0, 0);
    float y[8];
    load8(F + (size_t)srow[k] * NHID + joff, inScale, y);
    float py2 = 0.f, pxy = 0.f;
#pragma unroll
    for (int j = 0; j < 8; ++j) { py2 += y[j] * y[j]; pxy += x[j] * y[j]; }
    float y2 = wred32(py2);
    float xy = wred32(pxy);
    float den = fmaxf(1.f + 2.f * xy + x2 * y2, EPSF);
    float ca = (1.f + 2.f * xy + y2) / den;
    float cb = (1.f - x2) / den;
    float pn = 0.f;
#pragma unroll
    for (int j = 0; j < 8; ++j) { x[j] = ca * x[j] + cb * y[j]; pn += x[j] * x[j]; }
    x2 = wred32(pn);
  }

  float* orow = OUT + (size_t)node * NHID + joff;
  if (!doAct) {
    store8(orow, x, 1.f);
  } else {
    float b[8];
    load8(bias + joff, 1.f, b);
    float pb2 = 0.f, pxy = 0.f;
#pragma unroll
    for (int j = 0; j < 8; ++j) { pb2 += b[j] * b[j]; pxy += x[j] * b[j]; }
    float y2 = wred32(pb2);
    float xy = 0.25f * wred32(pxy);   // rst = 0.25*x
    float x2r = 0.0625f * x2;
    float den = fmaxf(1.f + 2.f * xy + x2r * y2, EPSF);
    float ca = (1.f + 2.f * xy + y2) / den;
    float cb = (1.f - x2r) / den;
    float zn2 = ca * ca * x2r + 2.f * ca * cb * xy + cb * cb * y2;
    float zn = fminf(fmaxf(sqrtf(zn2), EPSF), MAXNF);
    float us = atanhf(zn) / zn;
    float u[8];
    float pv = 0.f;
#pragma unroll
    for (int j = 0; j < 8; ++j) {
      float z = ca * 0.25f * x[j] + cb * b[j];
      u[j] = fmaxf(us * z, 0.f);
      pv += u[j] * u[j];
    }
    float vn = fmaxf(sqrtf(wred32(pv)), EPSF);
    float fs = tanhf(vn) / vn;
    store8(orow, u, fs);
  }
}

// ---------------------------------------------------------------------------
// Final: gather 64 rows of H (65536x256), mx = h @ Wl^T (64x64) via WMMA,
// mobius_matvec scale + mobius_add(bl) -> OUT (64x64). Single block.
// ---------------------------------------------------------------------------
__global__ __launch_bounds__(256) void k_final(const float* __restrict__ H,
                                               const float* __restrict__ Wl,
                                               const float* __restrict__ bl,
                                               const int* __restrict__ tf,
                                               float* __restrict__ OUT) {
  __shared__ int ridx[64];
  __shared__ float smx[64][72];
  __shared__ float part[64][4];
  __shared__ float rx2[64], rm2[64], rdot[64];
  __shared__ float sb2;

  const int t = threadIdx.x;
  if (t < 64) ridx[t] = tf[t] + t * 1024;  // n_per = 65536/64
  if (t == 0) {
    float s = 0.f;
    for (int j = 0; j < 64; ++j) s += bl[j] * bl[j];
    sb2 = s;
  }
  __syncthreads();

  const int wave = t >> 5, lane = t & 31;
  const int l15 = lane & 15;
  const int kk = (lane >> 4) << 1;
#pragma unroll
  for (int i = 0; i < 2; ++i) {
    const int tile = wave * 2 + i;         // 16 tiles over 8 waves
    const int tm = tile >> 2, tn = tile & 3;
    const float* Ar = H + (size_t)ridx[tm * 16 + l15] * NHID;
    const float* Br = Wl + (size_t)(tn * 16 + l15) * NHID;  // B[k][n] = Wl[n][k]
    v8f c = {};
    for (int k0 = 0; k0 < NHID; k0 += 4) {
      v2f a, b;
      a.x = Ar[k0 + kk]; a.y = Ar[k0 + kk + 1];
      b.x = Br[k0 + kk]; b.y = Br[k0 + kk + 1];
      c = wmma_f32x4(a, b, c);
    }
    const int n = tn * 16 + l15;
    const int mb = tm * 16 + ((lane >> 4) ? 8 : 0);
#pragma unroll
    for (int v = 0; v < 8; ++v) smx[mb + v][n] = c[v];
  }
  __syncthreads();

  const int r = t >> 2, p = t & 3;  // 4 threads per row
  float sx = 0.f;
  {
    const float* hr = H + (size_t)ridx[r] * NHID + p * 64;
    for (int j = 0; j < 64; j += 4) {
      float4 v = *(const float4*)(hr + j);
      sx += v.x * v.x + v.y * v.y + v.z * v.z + v.w * v.w;
    }
  }
  float sm = 0.f, sd = 0.f;
#pragma unroll
  for (int j = 0; j < 16; ++j) {
    float m = smx[r][p * 16 + j];
    sm += m * m;
    sd += m * bl[p * 16 + j];
  }
  part[r][p] = sx;
  __syncthreads();
  if (t < 64) rx2[t] = part[t][0] + part[t][1] + part[t][2] + part[t][3];
  __syncthreads();
  part[r][p] = sm;
  __syncthreads();
  if (t < 64) rm2[t] = part[t][0] + part[t][1] + part[t][2] + part[t][3];
  __syncthreads();
  part[r][p] = sd;
  __syncthreads();
  if (t < 64) rdot[t] = part[t][0] + part[t][1] + part[t][2] + part[t][3];
  __syncthreads();

  const float s = hyp_scale(rx2[r], rm2[r]);  // mm = s * mx_raw
  const float x2r = s * s * rm2[r];
  const float xy = s * rdot[r];
  const float y2 = sb2;
  const float den = fmaxf(1.f + 2.f * xy + x2r * y2, EPSF);
  const float ca = (1.f + 2.f * xy + y2) / den;
  const float cb = (1.f - x2r) / den;
  const float acs = ca * s;
#pragma unroll
  for (int j = 0; j < 16; ++j)
    OUT[r * 64 + p * 16 + j] = acs * smx[r][p * 16 + j] + cb * bl[p * 16 + j];
}

extern "C" void kernel_launch(void* const* d_in, const int* in_sizes, int n_in,
                              void* d_out, int out_size, void* d_ws,
                              size_t ws_size, hipStream_t stream) {
  const float* features = (const float*)d_in[0];  // 65536 x 512
  const float* W1 = (const float*)d_in[1];        // 512 x 256
  const float* b1 = (const float*)d_in[2];        // 256
  const float* W2 = (const float*)d_in[3];        // 256 x 256
  const float* b2 = (const float*)d_in[4];        // 256
  const float* Wl = (const float*)d_in[5];        // 64 x 256
  const float* bl = (const float*)d_in[6];        // 64
  const int* src_idx = (const int*)d_in[7];       // 65536 x 16
  const int* to_fetch = (const int*)d_in[8];      // 64
  float* out = (float*)d_out;                     // 64 x 64

  float* buf0 = (float*)d_ws;                     // 65536*256 f32
  float* buf1 = buf0 + (size_t)65536 * 256;       // 65536*256 f32

  // Layer 1: M1 = mobius_matvec(0.25*features, W1)
  k_mm<512, false><<<2048, 256, 0, stream>>>(const_cast<float*>(features), W1,
                                             b1, buf0, 0.25f);
  // Layer 1: aggregate(M1) -> *0.25 -> mobius_add(b1) -> act -> H1
  k_agg<<<8192, 256, 0, stream>>>(buf0, src_idx, b1, buf1, 1.0f, 1);
  // Layer 2: aggregate(0.25*H1) -> AG2
  k_agg<<<8192, 256, 0, stream>>>(buf1, src_idx, b2, buf0, 0.25f, 0);
  // Layer 2: mobius_matvec(AG2, W2) -> *0.25 -> mobius_add(b2) -> act (in-place)
  k_mm<256, true><<<2048, 256, 0, stream>>>(buf0, W2, b2, buf0, 1.0f);
  // Final: gather 64 rows, mobius_matvec(h, Wl^T), mobius_add(bl)
  k_final<<<1, 256, 0, stream>>>(buf0, Wl, bl, to_fetch, out);
}